// PDEEval_64476049048235
// MI455X (gfx1250) — compile-verified
//
#include <hip/hip_runtime.h>
#include <stdint.h>

// Problem constants (from reference)
#define BB      32
#define HH      512
#define WW      512
#define NPIX    (HH * WW)          // 262144
#define DT_C    0.2f
#define T_STEPS 16
#define ROWS    16                 // output rows per block
#define TILES   (HH / ROWS)        // 32 tiles per sample
#define NPART   TILES              // partials per sample == tiles per sample

// ---------------------------------------------------------------------------
// Pass 1: deterministic per-block partial sums of relu(x) (32 partials/sample)
// grid = BB*32 blocks of 256; each block owns 8192 contiguous floats,
// streamed as float4 (global_load_b128).
// ---------------------------------------------------------------------------
__global__ __launch_bounds__(256) void reduce_relu(const float* __restrict__ x,
                                                   float* __restrict__ partials) {
    int b   = blockIdx.x >> 5;   // sample
    int blk = blockIdx.x & 31;   // chunk within sample
    const float4* p4 = (const float4*)(x + (size_t)b * NPIX + (size_t)blk * 8192);

    float s = 0.f;
#pragma unroll
    for (int i = 0; i < 8; ++i) {
        float4 v = p4[threadIdx.x + i * 256];
        s += fmaxf(v.x, 0.f) + fmaxf(v.y, 0.f) + fmaxf(v.z, 0.f) + fmaxf(v.w, 0.f);
    }

    // wave32 reduction, then cross-wave via LDS
    for (int o = 16; o > 0; o >>= 1) s += __shfl_xor(s, o, 32);
    __shared__ float red[8];
    int lane = threadIdx.x & 31, wv = threadIdx.x >> 5;
    if (lane == 0) red[wv] = s;
    __syncthreads();
    if (threadIdx.x == 0) {
        float t = 0.f;
#pragma unroll
        for (int w = 0; w < 8; ++w) t += red[w];
        partials[blockIdx.x] = t;   // deterministic: one writer per slot
    }
}

// ---------------------------------------------------------------------------
// One Euler step for a 16-row x 512-col tile, with the per-sample D
// (sigmoid of mean concentration) recomputed in-block from the previous
// step's partials (uniform scalar loads -> SMEM path, no extra kernel).
//
// Rows [row0-1, row0+16] (periodic) are staged into LDS with CDNA5 async
// global->LDS B128 transfers (ASYNCcnt), then the 5-point stencil is served
// entirely from LDS. Emits this tile's partial sum of the NEW field into
// writeP for the next step's D.
// ---------------------------------------------------------------------------
__global__ __launch_bounds__(256) void step_kernel(const float* __restrict__ src,
                                                   float* __restrict__ dst,
                                                   const float* __restrict__ readP,
                                                   float* __restrict__ writeP,
                                                   const float* __restrict__ d_coef,
                                                   const int* __restrict__ tvec,
                                                   int step) {
    __shared__ float smem[(ROWS + 2) * WW];   // 18 * 512 * 4B = 36 KB
    __shared__ float red[8];

    int b    = blockIdx.x >> 5;          // sample
    int tile = blockIdx.x & 31;          // 32 tiles per sample
    int row0 = tile * ROWS;
    int tid  = threadIdx.x;
    const float* base = src + (size_t)b * NPIX;

    // ---- async stage: 18 rows * 512 floats = 2304 x 16B chunks, 9 per thread
#pragma unroll
    for (int it = 0; it < 9; ++it) {
        int chunk = it * 256 + tid;            // 0..2303
        int r     = chunk >> 7;                // LDS row (128 chunks per row)
        int c4    = chunk & 127;               // 16-byte chunk within row
        int gr    = (row0 - 1 + r + HH) & (HH - 1);   // periodic row wrap
        const float* gp = base + (size_t)gr * WW + (size_t)(c4 * 4);
        // generic LDS pointer: low 32 bits == LDS byte address (ISA 10.2)
        unsigned lo = (unsigned)(uintptr_t)(&smem[r * WW + c4 * 4]);
        asm volatile("global_load_async_to_lds_b128 %0, %1, off"
                     :: "v"(lo), "v"(gp) : "memory");
    }

    // ---- while the DMA-style async copies fly, compute this sample's D
    // (all operands uniform across the block -> scalar loads)
    float s = 0.f;
#pragma unroll
    for (int i = 0; i < NPART; ++i) s += readP[b * NPART + i];
    float mean = s * (1.0f / (float)NPIX);
    float sig  = 1.0f / (1.0f + __expf(-mean));
    float Dm   = d_coef[0] * sig * DT_C * ((tvec[b] > step) ? 1.0f : 0.0f);

    asm volatile("s_wait_asynccnt 0x0" ::: "memory");
    __syncthreads();

    // ---- 5-point stencil from LDS
    float* out = dst + (size_t)b * NPIX + (size_t)row0 * WW;
    int col0 = tid * 2;                                     // 2 adjacent cols/thread

    float acc = 0.f;
#pragma unroll
    for (int r = 0; r < ROWS; ++r) {
        float2 v;
        float* res = &v.x;
#pragma unroll
        for (int cc = 0; cc < 2; ++cc) {
            int col  = col0 + cc;
            float yc = smem[(r + 1) * WW + col];
            float cm = fmaxf(yc, 0.f);
            float cu = fmaxf(smem[ r      * WW + col], 0.f);
            float cd = fmaxf(smem[(r + 2) * WW + col], 0.f);
            float cl = fmaxf(smem[(r + 1) * WW + ((col + WW - 1) & (WW - 1))], 0.f);
            float cr = fmaxf(smem[(r + 1) * WW + ((col + 1)      & (WW - 1))], 0.f);
            float lap = cu + cd + cl + cr - 4.0f * cm;
            float yn  = fminf(fmaxf(yc + Dm * lap, 0.0f), 1.0f);
            res[cc] = yn;
            acc += yn;
        }
        *(float2*)(out + r * WW + col0) = v;   // global_store_b64
    }

    // ---- deterministic tile partial sum of the new field (for next step's D)
    for (int o = 16; o > 0; o >>= 1) acc += __shfl_xor(acc, o, 32);
    int lane = tid & 31, wv = tid >> 5;
    if (lane == 0) red[wv] = acc;
    __syncthreads();
    if (tid == 0) {
        float s2 = 0.f;
#pragma unroll
        for (int w = 0; w < 8; ++w) s2 += red[w];
        writeP[blockIdx.x] = s2;               // one writer per slot
    }
}

// ---------------------------------------------------------------------------
// Host-side launch sequence (graph-capturable, deterministic, 17 launches)
// ws layout: [ A: 32MB ping buffer | PA: 1024 f32 | PB: 1024 f32 ]
// ---------------------------------------------------------------------------
extern "C" void kernel_launch(void* const* d_in, const int* in_sizes, int n_in,
                              void* d_out, int out_size, void* d_ws, size_t ws_size,
                              hipStream_t stream) {
    const float* x      = (const float*)d_in[0];
    const int*   t      = (const int*)  d_in[1];
    const float* d_coef = (const float*)d_in[2];
    float* out = (float*)d_out;

    float* A  = (float*)d_ws;
    float* PA = (float*)((char*)d_ws + (size_t)BB * NPIX * sizeof(float));
    float* PB = PA + BB * NPART;

    // initial per-sample partial sums of relu(x) -> PA
    reduce_relu<<<BB * NPART, 256, 0, stream>>>(x, PA);

    const float* src = x;
    for (int s = 0; s < T_STEPS; ++s) {
        float* dst    = (s & 1) ? out : A;        // s=0 -> A, odd -> out; s=15 -> d_out
        float* readP  = (s & 1) ? PB  : PA;       // partials of field entering step s
        float* writeP = (s & 1) ? PA  : PB;       // partials of field leaving step s
        step_kernel<<<BB * TILES, 256, 0, stream>>>(src, dst, readP, writeP,
                                                    d_coef, t, s);
        src = dst;
    }
}